// NonstationaryGpModule_28630251995257
// MI455X (gfx1250) — compile-verified
//
#include <hip/hip_runtime.h>
#include <math.h>

#define Nn 1024
#define Mm 16384
#define Bb 2
#define MT 64      // test columns per workgroup in predict kernel
#define KTS 65     // padded LDS row stride for the K tile (bank-conflict avoidance)

typedef __attribute__((ext_vector_type(2))) float v2f;
typedef __attribute__((ext_vector_type(8))) float v8f;

#define SQRT3F 1.7320508075688772f
#define LOG2PIF 1.8378770664093453f

// Fused prob_product (quad + constant) + matern 3/2, matching the reference.
// p1/p2 layout: x, y, e00, e11, e01, det^0.25
__device__ __forceinline__ float kern_pair(float x1, float y1, float a1, float b1, float c1, float q1,
                                           float x2, float y2, float a2, float b2, float c2, float q2,
                                           float scale) {
    float s00 = a1 + a2, s11 = b1 + b2, s01 = c1 + c2;
    float sdet = s00 * s11 - s01 * s01;
    float d0 = x1 - x2, d1 = y1 - y2;
    float Q = 0.5f * (s11 * d0 * d0 - 2.0f * s01 * d0 * d1 + s00 * d1 * d1) / sdet;
    float C = 2.0f * q1 * q2 * rsqrtf(fmaxf(sdet, 1e-5f));
    float t = SQRT3F * sqrtf(fmaxf(Q, 1e-5f));
    return (1.0f + t) * __expf(-t) * C * scale;
}

// Stage 1: per-point derived params (x,y,a,b,c,det^0.25) for train & test, plus
// the diagonal kernel K_mm_diag for test points.
__global__ void prep_kernel(const float* __restrict__ ct, const float* __restrict__ kpt,
                            const float* __restrict__ cs, const float* __restrict__ kps,
                            const float* __restrict__ scale_param,
                            float* __restrict__ trainP, float* __restrict__ testP,
                            float* __restrict__ Kmm) {
    int gid = blockIdx.x * blockDim.x + threadIdx.x;
    float scale = __expf(scale_param[0]);
    if (gid < Bb * Mm) {
        float a = kps[gid * 3 + 0], b = kps[gid * 3 + 1], c = kps[gid * 3 + 2];
        float det = a * b - c * c;
        testP[gid * 6 + 0] = cs[gid * 2 + 0];
        testP[gid * 6 + 1] = cs[gid * 2 + 1];
        testP[gid * 6 + 2] = a;
        testP[gid * 6 + 3] = b;
        testP[gid * 6 + 4] = c;
        testP[gid * 6 + 5] = sqrtf(sqrtf(det));
        // diag kernel: C = 2*sqrt(det)/safe_sqrt(4 det); matern(Q=0) uses clip(0,1e-5)
        float t0 = SQRT3F * sqrtf(1e-5f);
        float m0 = (1.0f + t0) * __expf(-t0);
        float C = 2.0f * sqrtf(det) * rsqrtf(fmaxf(4.0f * det, 1e-5f));
        Kmm[gid] = C * m0 * scale;
    }
    if (gid < Bb * Nn) {
        float a = kpt[gid * 3 + 0], b = kpt[gid * 3 + 1], c = kpt[gid * 3 + 2];
        float det = a * b - c * c;
        trainP[gid * 6 + 0] = ct[gid * 2 + 0];
        trainP[gid * 6 + 1] = ct[gid * 2 + 1];
        trainP[gid * 6 + 2] = a;
        trainP[gid * 6 + 3] = b;
        trainP[gid * 6 + 4] = c;
        trainP[gid * 6 + 5] = sqrtf(sqrtf(det));
    }
}

// Stage 2: A = K_nn + diag(var)
__global__ void buildA_kernel(const float* __restrict__ trainP, const float* __restrict__ var,
                              const float* __restrict__ scale_param, float* __restrict__ A) {
    int gid = blockIdx.x * blockDim.x + threadIdx.x;   // Bb*Nn*Nn = 2^21, fits int
    int b = gid >> 20;
    int i = (gid >> 10) & (Nn - 1);
    int j = gid & (Nn - 1);
    const float* p1 = trainP + (size_t)(b * Nn + i) * 6;
    const float* p2 = trainP + (size_t)(b * Nn + j) * 6;
    float scale = __expf(scale_param[0]);
    float v = kern_pair(p1[0], p1[1], p1[2], p1[3], p1[4], p1[5],
                        p2[0], p2[1], p2[2], p2[3], p2[4], p2[5], scale);
    if (i == j) v += var[b * Nn + i];
    A[(size_t)gid] = v;
}

// Stage 3: in-place right-looking Cholesky, one workgroup per batch.
// Current column cached in LDS; also produces sum(log(diag(L))).
__global__ __launch_bounds__(1024) void chol_kernel(float* __restrict__ Abase,
                                                    float* __restrict__ logdet) {
    int b = blockIdx.x;
    float* A = Abase + (size_t)b * Nn * Nn;
    __shared__ float col[Nn];
    __shared__ float dinv_sh;
    int tid = threadIdx.x;
    for (int k = 0; k < Nn; k++) {
        if (tid == 0) {
            float d = sqrtf(A[(size_t)k * Nn + k]);
            A[(size_t)k * Nn + k] = d;
            dinv_sh = 1.0f / d;
        }
        __syncthreads();
        float dinv = dinv_sh;
        for (int i = k + 1 + tid; i < Nn; i += 1024) {
            float v = A[(size_t)i * Nn + k] * dinv;
            A[(size_t)i * Nn + k] = v;
            col[i] = v;
        }
        __syncthreads();
        for (int i = k + 1 + tid; i < Nn; i += 1024) {
            float ci = col[i];
            float* row = A + (size_t)i * Nn;
            for (int j = k + 1; j <= i; j++) row[j] -= ci * col[j];
        }
        __syncthreads();
    }
    // logdet reduction
    float ld = __logf(A[(size_t)tid * Nn + tid]);
    col[tid] = ld;
    __syncthreads();
    for (int s = 512; s > 0; s >>= 1) {
        if (tid < s) col[tid] += col[tid + s];
        __syncthreads();
    }
    if (tid == 0) logdet[b] = col[0];
}

// Stage 4: z = L^{-1}(y - mean); nlml = 0.5||z||^2 + logdet + 0.5 N log(2pi)
__global__ __launch_bounds__(256) void zsolve_kernel(const float* __restrict__ Lbase,
                                                     const float* __restrict__ y,
                                                     const float* __restrict__ mean,
                                                     const float* __restrict__ logdet,
                                                     float* __restrict__ zbase,
                                                     float* __restrict__ nlml_out) {
    int b = blockIdx.x;
    const float* L = Lbase + (size_t)b * Nn * Nn;
    __shared__ float zsh[Nn];
    __shared__ float red[256];
    int tid = threadIdx.x;
    float mb = mean[b];
    for (int i = 0; i < Nn; i++) {
        const float* row = L + (size_t)i * Nn;
        float s = 0.0f;
        for (int j = tid; j < i; j += 256) s += row[j] * zsh[j];
        red[tid] = s;
        __syncthreads();
        for (int st = 128; st > 0; st >>= 1) {
            if (tid < st) red[tid] += red[tid + st];
            __syncthreads();
        }
        if (tid == 0) zsh[i] = (y[b * Nn + i] - mb - red[0]) / row[i];
        __syncthreads();
    }
    float s2 = 0.0f;
    for (int i = tid; i < Nn; i += 256) {
        float z = zsh[i];
        zbase[b * Nn + i] = z;
        s2 += z * z;
    }
    red[tid] = s2;
    __syncthreads();
    for (int st = 128; st > 0; st >>= 1) {
        if (tid < st) red[tid] += red[tid + st];
        __syncthreads();
    }
    if (tid == 0) nlml_out[b] = 0.5f * red[0] + logdet[b] + 0.5f * Nn * LOG2PIF;
}

// Stage 5: explicit lower-triangular inverse Linv = L^{-1}.
// One wave32 per column; x kept in LDS; shfl_xor dot reductions.
__global__ __launch_bounds__(256) void trinv_kernel(const float* __restrict__ Lbase,
                                                    float* __restrict__ Linvbase) {
    int b = blockIdx.y;
    int wave = threadIdx.x >> 5;
    int lane = threadIdx.x & 31;
    int j = blockIdx.x * 8 + wave;
    const float* L = Lbase + (size_t)b * Nn * Nn;
    float* Li = Linvbase + (size_t)b * Nn * Nn;
    __shared__ float xsh[8 * Nn];
    float* x = xsh + wave * Nn;
    // zero the strictly-upper part of this column (predict kernel reads full 16x16 blocks)
    for (int i = lane; i < j; i += 32) Li[(size_t)i * Nn + j] = 0.0f;
    float djj = 1.0f / L[(size_t)j * Nn + j];
    if (lane == 0) {
        x[j] = djj;
        Li[(size_t)j * Nn + j] = djj;
    }
    for (int i = j + 1; i < Nn; i++) {
        const float* row = L + (size_t)i * Nn;
        float s = 0.0f;
        for (int k = j + lane; k < i; k += 32) s += row[k] * x[k];
        s += __shfl_xor(s, 16);
        s += __shfl_xor(s, 8);
        s += __shfl_xor(s, 4);
        s += __shfl_xor(s, 2);
        s += __shfl_xor(s, 1);
        float xi = -s / row[i];
        if (lane == 0) {
            x[i] = xi;
            Li[(size_t)i * Nn + j] = xi;
        }
    }
}

// Stage 6 (the big one): fused K_nm-tile generation + w = Linv @ K via fp32 WMMA
// + on-the-fly contraction into pred_mean / pred_var. 4 waves x 16 columns each.
__global__ __launch_bounds__(128) void predict_kernel(const float* __restrict__ Linvbase,
                                                      const float* __restrict__ trainPg,
                                                      const float* __restrict__ testPg,
                                                      const float* __restrict__ Kmm,
                                                      const float* __restrict__ zbase,
                                                      const float* __restrict__ meang,
                                                      const float* __restrict__ scale_param,
                                                      float* __restrict__ out_mean,
                                                      float* __restrict__ out_var) {
    extern __shared__ float smem[];
    float* Kt  = smem;                  // Nn * KTS  (K tile, padded)
    float* zsh = smem + Nn * KTS;       // Nn
    float* tp  = zsh + Nn;              // Nn * 6 train params
    int b = blockIdx.y;
    int colBase = blockIdx.x * MT;
    int tid = threadIdx.x;
    float scale = __expf(scale_param[0]);

    for (int idx = tid; idx < Nn; idx += 128) zsh[idx] = zbase[b * Nn + idx];
    for (int idx = tid; idx < Nn * 6; idx += 128) tp[idx] = trainPg[(size_t)b * Nn * 6 + idx];
    __syncthreads();

    // Build the 1024 x 64 K tile in LDS (never touches HBM).
    {
        int c = tid & 63;
        int h = tid >> 6;   // uniform per wave
        const float* q = testPg + (size_t)(b * Mm + colBase + c) * 6;
        float x2 = q[0], y2 = q[1], a2 = q[2], b2 = q[3], c2 = q[4], q2 = q[5];
        for (int n = h; n < Nn; n += 2) {
            const float* p = tp + n * 6;
            Kt[n * KTS + c] = kern_pair(p[0], p[1], p[2], p[3], p[4], p[5],
                                        x2, y2, a2, b2, c2, q2, scale);
        }
    }
    __syncthreads();

    int wave = tid >> 5, lane = tid & 31;
    int r = lane & 15, hi = lane >> 4;
    int wcol = wave * 16;
    const float* Li = Linvbase + (size_t)b * Nn * Nn;

    float meanAcc = 0.0f, varAcc = 0.0f;
    for (int rb = 0; rb < Nn / 16; rb++) {
        v8f acc = {};
        const float* arow = Li + (size_t)(rb * 16 + r) * Nn;
        for (int j = 0; j <= rb; j++) {
            int kb = j * 16;
#pragma unroll
            for (int k0 = 0; k0 < 16; k0 += 4) {
                int kk = kb + k0 + 2 * hi;
                // A (16x4 fp32): lane r holds K = {kk, kk+1} -> one 8B load from Linv (L2-resident)
                v2f a = *(const v2f*)(arow + kk);
                // B (4x16 fp32): lane holds col r of rows K = {kk, kk+1} from LDS
                v2f bb;
                bb.x = Kt[kk * KTS + wcol + r];
                bb.y = Kt[(kk + 1) * KTS + wcol + r];
                acc = __builtin_amdgcn_wmma_f32_16x16x4_f32(false, a, false, bb,
                                                            (short)0, acc, false, false);
            }
        }
        // C layout: VGPR v holds row (v + 8*hi), col r. Contract immediately; w never stored.
        const float* zb = zsh + rb * 16 + 8 * hi;
#pragma unroll
        for (int v = 0; v < 8; v++) {
            float cv = acc[v];
            meanAcc += zb[v] * cv;
            varAcc  += cv * cv;
        }
    }
    meanAcc += __shfl_xor(meanAcc, 16);
    varAcc  += __shfl_xor(varAcc, 16);
    if (lane < 16) {
        int m = colBase + wcol + lane;
        out_mean[b * Mm + m] = meang[b] + meanAcc;
        out_var[b * Mm + m]  = Kmm[b * Mm + m] - varAcc;
    }
}

extern "C" void kernel_launch(void* const* d_in, const int* in_sizes, int n_in,
                              void* d_out, int out_size, void* d_ws, size_t ws_size,
                              hipStream_t stream) {
    const float* coords_train = (const float*)d_in[0];
    const float* kp_train     = (const float*)d_in[1];
    const float* coords_test  = (const float*)d_in[2];
    const float* kp_test      = (const float*)d_in[3];
    const float* y_train      = (const float*)d_in[4];
    const float* var          = (const float*)d_in[5];
    const float* mean         = (const float*)d_in[6];
    const float* scale_param  = (const float*)d_in[7];

    float* ws = (float*)d_ws;
    size_t off = 0;
    float* A      = ws + off; off += (size_t)Bb * Nn * Nn;   // K_nn+diag(var) -> L (in place)
    float* Linv   = ws + off; off += (size_t)Bb * Nn * Nn;   // explicit L^{-1}
    float* trainP = ws + off; off += (size_t)Bb * Nn * 6;
    float* testP  = ws + off; off += (size_t)Bb * Mm * 6;
    float* Kmm    = ws + off; off += (size_t)Bb * Mm;
    float* z      = ws + off; off += (size_t)Bb * Nn;
    float* logdet = ws + off; off += (size_t)Bb;

    float* out_mean = (float*)d_out;                 // [B*M]
    float* out_var  = out_mean + (size_t)Bb * Mm;    // [B*M]
    float* nlml     = out_var + (size_t)Bb * Mm;     // [B]

    prep_kernel<<<(Bb * Mm + 255) / 256, 256, 0, stream>>>(
        coords_train, kp_train, coords_test, kp_test, scale_param, trainP, testP, Kmm);
    buildA_kernel<<<(Bb * Nn * Nn) / 256, 256, 0, stream>>>(trainP, var, scale_param, A);
    chol_kernel<<<Bb, 1024, 0, stream>>>(A, logdet);
    zsolve_kernel<<<Bb, 256, 0, stream>>>(A, y_train, mean, logdet, z, nlml);
    trinv_kernel<<<dim3(Nn / 8, Bb), 256, 0, stream>>>(A, Linv);

    size_t smem_bytes = (size_t)(Nn * KTS + Nn + Nn * 6) * sizeof(float);
    predict_kernel<<<dim3(Mm / MT, Bb), 128, smem_bytes, stream>>>(
        Linv, trainP, testP, Kmm, z, mean, scale_param, out_mean, out_var);
}